// LQR_6957847019950
// MI455X (gfx1250) — compile-verified
//
#include <hip/hip_runtime.h>

// ---------------------------------------------------------------------------
// Batched LQR (Riccati backward + rollout) for MI455X / gfx1250.
// One wave32 == one batch problem. fp32 WMMA (V_WMMA_F32_16X16X4_F32) for all
// tile matmuls. F=[A|B] fragments hoisted to registers for the whole t-loop.
// Q_t double-buffered in LDS via GLOBAL_LOAD_ASYNC_TO_LDS_B128 (ASYNCcnt).
// ---------------------------------------------------------------------------

typedef __attribute__((ext_vector_type(2))) float v2f;
typedef __attribute__((ext_vector_type(8))) float v8f;
typedef __attribute__((ext_vector_type(4))) int v4i;

#define NS  32
#define NC  16
#define NSC 48

#define GLOBAL_AS __attribute__((address_space(1)))
#define LDS_AS    __attribute__((address_space(3)))

__device__ __forceinline__ v8f vzero8() {
  v8f z;
#pragma unroll
  for (int i = 0; i < 8; ++i) z[i] = 0.0f;
  return z;
}

// A-fragment (16x4 f32) from row-major src: A[m][k] = src[m*ld + k]
__device__ __forceinline__ v2f ld_fragA(const float* src, int ld) {
  const int l = threadIdx.x;
  const int m = l & 15;
  const int k = (l >> 4) << 1;
  v2f a;
  a.x = src[m * ld + k];
  a.y = src[m * ld + k + 1];
  return a;
}

// B-fragment (4x16 f32) from row-major src: B[k][n] = src[k*ld + n]
// (identical lane gather serves as A-fragment of src^T)
__device__ __forceinline__ v2f ld_fragB(const float* src, int ld) {
  const int l = threadIdx.x;
  const int n = l & 15;
  const int k = (l >> 4) << 1;
  v2f b;
  b.x = src[k * ld + n];
  b.y = src[(k + 1) * ld + n];
  return b;
}

// C/D tile (16x16 f32): VGPR r holds row (r + 8*(lane>>4)), col lane&15
__device__ __forceinline__ v8f ld_tileC(const float* src, int ld) {
  const int l = threadIdx.x;
  const int n = l & 15;
  const int m0 = (l >> 4) << 3;
  v8f c;
#pragma unroll
  for (int r = 0; r < 8; ++r) c[r] = src[(m0 + r) * ld + n];
  return c;
}

__device__ __forceinline__ void st_tileC(float* dst, int ld, v8f c) {
  const int l = threadIdx.x;
  const int n = l & 15;
  const int m0 = (l >> 4) << 3;
#pragma unroll
  for (int r = 0; r < 8; ++r) dst[(m0 + r) * ld + n] = c[r];
}

__device__ __forceinline__ v8f wmma4(v8f acc, v2f a, v2f b) {
  return __builtin_amdgcn_wmma_f32_16x16x4_f32(false, a, false, b, (short)0,
                                               acc, false, false);
}

// acc += A(16xK, row-major a, ld la) * B(Kx16, row-major b, ld lb)
template <int K>
__device__ __forceinline__ v8f mm_nn(v8f acc, const float* a, int la,
                                     const float* b, int lb) {
#pragma unroll
  for (int k = 0; k < K; k += 4)
    acc = wmma4(acc, ld_fragA(a + k, la), ld_fragB(b + k * lb, lb));
  return acc;
}

// ---- async global -> LDS copy of one Q_t tile (9216 B), double-buffered ----
__device__ __forceinline__ void q_async_load(const float* g, float* lds_dst) {
  const int l = threadIdx.x;
#if __has_builtin(__builtin_amdgcn_global_load_async_to_lds_b128)
  GLOBAL_AS v4i* gp = (GLOBAL_AS v4i*)(g + 4 * l);
  LDS_AS v4i* lp = (LDS_AS v4i*)(lds_dst + 4 * l);
#pragma unroll
  for (int c = 0; c < (NSC * NSC) / (4 * 32); ++c)  // 18 x b128 per lane
    __builtin_amdgcn_global_load_async_to_lds_b128(gp + c * 32, lp + c * 32,
                                                   0, 0);
#else
  const float4* g4 = (const float4*)g;
  float4* l4 = (float4*)lds_dst;
  for (int i = l; i < (NSC * NSC) / 4; i += 32) l4[i] = g4[i];
#endif
}

__device__ __forceinline__ void q_async_wait() {
#if __has_builtin(__builtin_amdgcn_s_wait_asynccnt)
  __builtin_amdgcn_s_wait_asynccnt(0);
#elif __has_builtin(__builtin_amdgcn_global_load_async_to_lds_b128)
  asm volatile("s_wait_asynccnt 0x0" ::: "memory");
#endif
}

extern "C" __global__ __launch_bounds__(32)
void lqr_fused(const float* __restrict__ xinit, const float* __restrict__ Q,
               const float* __restrict__ p, const float* __restrict__ A,
               const float* __restrict__ B, const float* __restrict__ c1,
               float* __restrict__ xout, float* __restrict__ uout,
               float* __restrict__ cost, int T) {
  extern __shared__ float smem[];
  const int b = blockIdx.x;
  const int l = threadIdx.x;

  // ---- LDS partition -------------------------------------------------------
  float* sKs  = smem;                        // T*NC*NS
  float* sks  = sKs + (size_t)T * NC * NS;   // T*NC
  float* sF   = sks + (size_t)T * NC;        // NS*NSC   F = [A|B]
  float* sQ0  = sF + NS * NSC;               // NSC*NSC  Q/Qe double buffer 0
  float* sQ1  = sQ0 + NSC * NSC;             // NSC*NSC  Q/Qe double buffer 1
  float* sM   = sQ1 + NSC * NSC;             // NS*NSC   M = V*F
  float* sV   = sM + NS * NSC;               // NS*NS
  float* sAug = sV + NS * NS;                // NC*2NC   Gauss-Jordan
  float* sQi  = sAug + NC * 2 * NC;          // NC*NC    -(Quu^-1)
  float* sqe  = sQi + NC * NC;               // NSC
  float* sw   = sqe + NSC;                   // NS
  float* sv   = sw + NS;                     // NS
  float* skt  = sv + NS;                     // NC
  float* sc1  = skt + NC;                    // NS
  float* sx   = sc1 + NS;                    // NS
  float* su   = sx + NS;                     // NC
  float* sxu  = su + NC;                     // NSC

  // ---- preload F = [A|B], c1 ; init V=0, v=0 ------------------------------
  for (int i = l; i < NS * NS; i += 32) {
    int r = i >> 5, c = i & 31;
    sF[r * NSC + c] = A[(size_t)b * NS * NS + i];
  }
  for (int i = l; i < NS * NC; i += 32) {
    int r = i >> 4, c = i & 15;
    sF[r * NSC + NS + c] = B[(size_t)b * NS * NC + i];
  }
  if (l < NS) sc1[l] = c1[(size_t)b * NS + l];
  for (int i = l; i < NS * NS; i += 32) sV[i] = 0.0f;
  if (l < NS) sv[l] = 0.0f;
  __syncthreads();

  // ---- hoist all F fragments into registers (time-invariant) --------------
  // fB[blk][kk] serves BOTH as B-operand of V*F (block=column block of F)
  // and as A-operand of F^T*M (block=row block of F^T): same lane gather.
  v2f fB[3][8];
#pragma unroll
  for (int blk = 0; blk < 3; ++blk)
#pragma unroll
    for (int kk = 0; kk < 8; ++kk)
      fB[blk][kk] = ld_fragB(sF + blk * 16 + kk * 4 * NSC, NSC);

  // ---- prime Q double buffer with Q[T-1] ----------------------------------
  q_async_load(Q + ((size_t)b * T + (T - 1)) * (NSC * NSC),
               ((T - 1) & 1) ? sQ1 : sQ0);

  // ======================= backward Riccati pass ===========================
#pragma unroll 1
  for (int t = T - 1; t >= 0; --t) {
    float* cur = (t & 1) ? sQ1 : sQ0;   // holds Q_t, accumulated into Qe
    float* nxt = (t & 1) ? sQ0 : sQ1;
    const float* pg = p + ((size_t)b * T + t) * NSC;

    q_async_wait();                      // Q_t resident in `cur`
    __syncthreads();
    if (t > 0)                           // overlap next tile with this step
      q_async_load(Q + ((size_t)b * T + (t - 1)) * (NSC * NSC), nxt);

    // M = V * F   (32x48, 6 tiles, K=32) ; B operand from registers
#pragma unroll
    for (int i = 0; i < 2; ++i)
#pragma unroll
      for (int j = 0; j < 3; ++j) {
        v8f acc = vzero8();
#pragma unroll
        for (int kk = 0; kk < 8; ++kk)
          acc = wmma4(acc, ld_fragA(sV + i * 16 * NS + kk * 4, NS), fB[j][kk]);
        st_tileC(sM + i * 16 * NSC + j * 16, NSC, acc);
      }
    __syncthreads();

    // Qe = Q_t + F^T * M   (48x48, 9 tiles, K=32) ; A operand from registers
#pragma unroll
    for (int i = 0; i < 3; ++i)
#pragma unroll
      for (int j = 0; j < 3; ++j) {
        float* base = cur + i * 16 * NSC + j * 16;
        v8f acc = ld_tileC(base, NSC);
#pragma unroll
        for (int kk = 0; kk < 8; ++kk)
          acc = wmma4(acc, fB[i][kk], ld_fragB(sM + j * 16 + kk * 4 * NSC, NSC));
        st_tileC(base, NSC, acc);
      }

    // w = v + V*c1
    if (l < NS) {
      float s = sv[l];
      for (int k = 0; k < NS; ++k) s += sV[l * NS + k] * sc1[k];
      sw[l] = s;
    }
    __syncthreads();

    // qe = p_t + F^T * w
    for (int j = l; j < NSC; j += 32) {
      float s = pg[j];
      for (int k = 0; k < NS; ++k) s += sF[k * NSC + j] * sw[k];
      sqe[j] = s;
    }
    __syncthreads();

    // Gauss-Jordan inverse of Quu (SPD -> no pivoting); store -(Quu^-1)
    for (int i = l; i < NC * 2 * NC; i += 32) {
      int r = i >> 5, c = i & 31;
      sAug[i] = (c < NC) ? cur[(NS + r) * NSC + NS + c]
                         : ((c - NC) == r ? 1.0f : 0.0f);
    }
    __syncthreads();
#pragma unroll 1
    for (int k = 0; k < NC; ++k) {
      float pinv = 1.0f / sAug[k * 32 + k];
      float rk = sAug[k * 32 + l] * pinv;  // lane l owns column l
      __syncthreads();
      sAug[k * 32 + l] = rk;
#pragma unroll 1
      for (int r = 0; r < NC; ++r) {
        if (r != k) {
          float f = sAug[r * 32 + k];      // wave-wide load precedes store
          sAug[r * 32 + l] -= f * rk;
        }
      }
      __syncthreads();
    }
    for (int i = l; i < NC * NC; i += 32) {
      int r = i >> 4, c = i & 15;
      sQi[i] = -sAug[r * 32 + NC + c];
    }
    __syncthreads();

    // k_t = -(Quu^-1) * qu
    if (l < NC) {
      float s = 0.0f;
      for (int k = 0; k < NC; ++k) s += sQi[l * NC + k] * sqe[NS + k];
      skt[l] = s;
      sks[t * NC + l] = s;
    }
    // K_t = -(Quu^-1) * Qux   (16x32, 2 tiles, K=16)
    float* Kt = sKs + (size_t)t * NC * NS;
#pragma unroll
    for (int j = 0; j < 2; ++j) {
      v8f acc = vzero8();
      acc = mm_nn<NC>(acc, sQi, NC, cur + NS * NSC + j * 16, NSC);
      st_tileC(Kt + j * 16, NS, acc);
    }
    __syncthreads();

    // V = Qxx + Qxu*K_t  (Schur complement; remaining terms cancel exactly)
#pragma unroll
    for (int i = 0; i < 2; ++i)
#pragma unroll
      for (int j = 0; j < 2; ++j) {
        v8f acc = ld_tileC(cur + i * 16 * NSC + j * 16, NSC);
        acc = mm_nn<NC>(acc, cur + i * 16 * NSC + NS, NSC, Kt + j * 16, NS);
        st_tileC(sV + i * 16 * NS + j * 16, NS, acc);
      }
    // v = qx + Qxu*k_t
    if (l < NS) {
      float s = sqe[l];
      for (int k = 0; k < NC; ++k) s += cur[l * NSC + NS + k] * skt[k];
      sv[l] = s;
    }
    __syncthreads();
  }

  // ========================= forward rollout ===============================
  if (l < NS) sx[l] = xinit[(size_t)b * NS + l];
  float costacc = 0.0f;
  __syncthreads();
#pragma unroll 1
  for (int t = 0; t < T; ++t) {
    const float* Qg = Q + ((size_t)b * T + t) * (NSC * NSC);
    const float* pg = p + ((size_t)b * T + t) * NSC;
    const float* Kt = sKs + (size_t)t * NC * NS;
    if (t + 1 < T)
      __builtin_prefetch(Q + ((size_t)b * T + (t + 1)) * (NSC * NSC), 0, 1);

    if (l < NC) {
      float s = sks[t * NC + l];
      for (int k = 0; k < NS; ++k) s += Kt[l * NS + k] * sx[k];
      su[l] = s;
    }
    __syncthreads();
    if (l < NS) sxu[l] = sx[l];
    if (l < NC) sxu[NS + l] = su[l];

    if (l < NS) xout[(size_t)b * (T + 1) * NS + (size_t)t * NS + l] = sx[l];
    if (l < NC) uout[(size_t)b * T * NC + (size_t)t * NC + l] = su[l];
    __syncthreads();

    // cost_t = 0.5*xu^T Q xu + p.xu  (lane handles rows l and l+32)
    for (int r = l; r < NSC; r += 32) {
      const float4* row = (const float4*)(Qg + r * NSC);
      float s = 0.0f;
#pragma unroll
      for (int c4 = 0; c4 < NSC / 4; ++c4) {
        float4 q = row[c4];
        s += q.x * sxu[c4 * 4 + 0] + q.y * sxu[c4 * 4 + 1] +
             q.z * sxu[c4 * 4 + 2] + q.w * sxu[c4 * 4 + 3];
      }
      costacc += sxu[r] * (0.5f * s + pg[r]);
    }

    // x <- A x + B u + c1
    float xn = 0.0f;
    if (l < NS) {
      xn = sc1[l];
      for (int k = 0; k < NSC; ++k) xn += sF[l * NSC + k] * sxu[k];
    }
    __syncthreads();
    if (l < NS) sx[l] = xn;
    __syncthreads();
  }
  if (l < NS) xout[(size_t)b * (T + 1) * NS + (size_t)T * NS + l] = sx[l];

#pragma unroll
  for (int off = 16; off > 0; off >>= 1)
    costacc += __shfl_xor(costacc, off, 32);
  if (l == 0) cost[b] = costacc;
}

extern "C" void kernel_launch(void* const* d_in, const int* in_sizes, int n_in,
                              void* d_out, int out_size, void* d_ws,
                              size_t ws_size, hipStream_t stream) {
  const float* xinit = (const float*)d_in[0];
  const float* Q     = (const float*)d_in[1];
  const float* p     = (const float*)d_in[2];
  const float* A     = (const float*)d_in[3];
  const float* Bm    = (const float*)d_in[4];
  const float* c1    = (const float*)d_in[5];

  const int nb = in_sizes[0] / NS;                // 1024
  const int T  = in_sizes[1] / (nb * NSC * NSC);  // 64

  float* out  = (float*)d_out;
  float* xout = out;                               // nb*(T+1)*NS
  float* uout = xout + (size_t)nb * (T + 1) * NS;  // nb*T*NC
  float* cost = uout + (size_t)nb * T * NC;        // nb

  const size_t shmem =
      ((size_t)T * (NC * NS + NC) +           // K_t, k_t cache
       NS * NSC + 2 * NSC * NSC + NS * NSC +  // F, Q double buffer, M
       NS * NS + NC * 2 * NC + NC * NC +      // V, GJ aug, Quu^-1
       NSC + NS + NS + NC + NS + NS + NC + NSC) *
      sizeof(float);  // ~170 KB < 320 KB/WGP

  (void)hipFuncSetAttribute(reinterpret_cast<const void*>(lqr_fused),
                            hipFuncAttributeMaxDynamicSharedMemorySize,
                            (int)shmem);
  lqr_fused<<<dim3(nb), dim3(32), shmem, stream>>>(xinit, Q, p, A, Bm, c1,
                                                   xout, uout, cost, T);
}